// Deepseekv3MoE_206158430271
// MI455X (gfx1250) — compile-verified
//
#include <hip/hip_runtime.h>
#include <math.h>
#include <stdint.h>

#define TOKENS   512
#define HIDDEN   2048
#define INTER    1024
#define SH_INTER 2048
#define NEXP     32
#define TOPK     8
#define NGROUP   8
#define GSZ      (NEXP / NGROUP)   /* 4 experts per group */
#define TOPKG    4
#define RSF      2.5f

typedef float v2f __attribute__((ext_vector_type(2)));
typedef float v8f __attribute__((ext_vector_type(8)));

__device__ __forceinline__ v8f wmma4(v2f a, v2f b, v8f c) {
  // D = A(16x4 f32) * B(4x16 f32) + C(16x16 f32)
  return __builtin_amdgcn_wmma_f32_16x16x4_f32(false, a, false, b, (short)0, c,
                                               false, false);
}

__device__ __forceinline__ float silu(float g) {
  return g / (1.f + __expf(-g));
}

// 16-byte asynchronous global -> LDS copy (ASYNCcnt-tracked, ISA 15.18.3).
// lds is the wave-relative LDS byte offset (low 32 bits of the flat address).
__device__ __forceinline__ void async_copy16(unsigned lds, const float* gptr) {
  asm volatile("global_load_async_to_lds_b128 %0, %1, off"
               :: "v"(lds), "v"(gptr)
               : "memory");
}
__device__ __forceinline__ unsigned lds_off(const void* p) {
  return (unsigned)(uintptr_t)p;   // LDS aperture decode uses addr[31:0]
}
__device__ __forceinline__ void wait_async_le8() {
  asm volatile("s_wait_asynccnt 0x8" ::: "memory");
}
__device__ __forceinline__ void wait_async_le6() {
  asm volatile("s_wait_asynccnt 0x6" ::: "memory");
}
__device__ __forceinline__ void wait_async_le4() {
  asm volatile("s_wait_asynccnt 0x4" ::: "memory");
}
__device__ __forceinline__ void wait_async_le3() {
  asm volatile("s_wait_asynccnt 0x3" ::: "memory");
}
__device__ __forceinline__ void wait_async_0() {
  asm volatile("s_wait_asynccnt 0x0" ::: "memory");
}

// ---------------------------------------------------------------------------
// Kernel 1: gate logits + DeepSeek-V3 noaux-tc routing. One block per token.
// ---------------------------------------------------------------------------
__global__ __launch_bounds__(256) void gate_kernel(
    const float* __restrict__ x, const float* __restrict__ gw,
    const float* __restrict__ gbias, float* __restrict__ route) {
  __shared__ float logits[NEXP];
  __shared__ float swb[NEXP];
  __shared__ float sc[NEXP];
  __shared__ float gsc[NGROUP];
  __shared__ float masked[NEXP];
  __shared__ float selv[NEXP];
  __shared__ float ssum;

  const int t = blockIdx.x;
  const int tid = threadIdx.x;
  const int lane = tid & 31;
  const int wv = tid >> 5;
  const float* xt = x + (size_t)t * HIDDEN;

  // logits: 8 waves, each handles experts wv, wv+8, wv+16, wv+24
  for (int it = 0; it < 4; ++it) {
    const int e = wv + it * 8;
    const float* ge = gw + (size_t)e * HIDDEN;
    float acc = 0.f;
    for (int j = lane; j < HIDDEN; j += 32) acc += xt[j] * ge[j];
    for (int off = 16; off > 0; off >>= 1) acc += __shfl_xor(acc, off, 32);
    if (lane == 0) logits[e] = acc;
  }
  __syncthreads();

  if (tid < NEXP) {
    const float s = 1.f / (1.f + __expf(-logits[tid]));
    sc[tid] = s;
    swb[tid] = s + gbias[tid];
  }
  __syncthreads();

  // group score = sum of top-2 biased scores within each group of 4
  if (tid < NGROUP) {
    const float v0 = swb[tid * GSZ + 0], v1 = swb[tid * GSZ + 1];
    const float v2 = swb[tid * GSZ + 2], v3 = swb[tid * GSZ + 3];
    const float a01 = fmaxf(v0, v1), b01 = fminf(v0, v1);
    const float a23 = fmaxf(v2, v3), b23 = fminf(v2, v3);
    const float m1 = fmaxf(a01, a23);
    const float m2 = (a01 >= a23) ? fmaxf(a23, b01) : fmaxf(a01, b23);
    gsc[tid] = m1 + m2;
  }
  __syncthreads();

  // mask experts whose group is not among top-4 groups
  if (tid < NEXP) {
    const int g = tid / GSZ;
    const float mygs = gsc[g];
    int grank = 0;
    for (int j = 0; j < NGROUP; ++j) {
      const float o = gsc[j];
      grank += (o > mygs || (o == mygs && j < g)) ? 1 : 0;
    }
    masked[tid] = (grank < TOPKG) ? swb[tid] : -1e30f;
  }
  __syncthreads();

  // top-8 experts by masked biased score; select unbiased sigmoid score
  if (tid < NEXP) {
    const float mv = masked[tid];
    int rank = 0;
    for (int j = 0; j < NEXP; ++j) {
      const float o = masked[j];
      rank += (o > mv || (o == mv && j < tid)) ? 1 : 0;
    }
    selv[tid] = (rank < TOPK) ? sc[tid] : 0.f;
  }
  __syncthreads();

  if (tid == 0) {
    float s = 0.f;
    for (int j = 0; j < NEXP; ++j) s += selv[j];
    ssum = s;
  }
  __syncthreads();

  if (tid < NEXP)
    route[(size_t)t * NEXP + tid] = selv[tid] / (ssum + 1e-20f) * RSF;
}

// ---------------------------------------------------------------------------
// Kernel 2: deterministic per-expert compaction (token-order scan, no atomics)
// ---------------------------------------------------------------------------
__global__ __launch_bounds__(64) void compact_kernel(
    const float* __restrict__ route, int* __restrict__ list,
    float* __restrict__ wgt, int* __restrict__ counts,
    int* __restrict__ offs) {
  const int e = threadIdx.x;
  if (e < NEXP) {
    int c = 0;
    for (int t = 0; t < TOKENS; ++t) {
      const float w = route[(size_t)t * NEXP + e];
      if (w != 0.f) {
        list[e * TOKENS + c] = t;
        wgt[e * TOKENS + c] = w;
        ++c;
      }
    }
    counts[e] = c;
  }
  __syncthreads();
  if (e == 0) {
    int acc = 0;
    for (int j = 0; j < NEXP; ++j) {
      offs[j] = acc;
      acc += counts[j];
    }
  }
}

// ---------------------------------------------------------------------------
// Shared expert, stage 1: act = silu(x @ wg^T) * (x @ wu^T)   [512, 2048]
// Tile 64(M) x 32(N) x 32(K); 8 wave32s; triple-buffered async-LDS pipeline,
// one barrier per K-tile (issue at iter i writes buffer read at iter i-1,
// which all waves finished before barrier(i)).
// ---------------------------------------------------------------------------
__global__ __launch_bounds__(256) void shared_mlp1_kernel(
    const float* __restrict__ x, const float* __restrict__ wg,
    const float* __restrict__ wu, float* __restrict__ act) {
  __shared__ float xs[3][64][36];
  __shared__ float bg[3][32][36];
  __shared__ float bu[3][32][36];
  const int tid = threadIdx.x;
  const int lane = tid & 31, wv = tid >> 5;
  const int wm = wv & 3, wn = wv >> 2;
  const int half = lane >> 4, l16 = lane & 15;
  const int mBase = blockIdx.y * 64;
  const int nBase = blockIdx.x * 32;
  const int r0 = tid >> 3;        // 0..31
  const int c4 = (tid & 7) * 4;   // 0,4,..,28

  v8f cg = {};
  v8f cu = {};

  auto issue = [&](int buf, int k0) {
    async_copy16(lds_off(&xs[buf][r0][c4]),
                 x + (size_t)(mBase + r0) * HIDDEN + k0 + c4);
    async_copy16(lds_off(&xs[buf][r0 + 32][c4]),
                 x + (size_t)(mBase + r0 + 32) * HIDDEN + k0 + c4);
    async_copy16(lds_off(&bg[buf][r0][c4]),
                 wg + (size_t)(nBase + r0) * HIDDEN + k0 + c4);
    async_copy16(lds_off(&bu[buf][r0][c4]),
                 wu + (size_t)(nBase + r0) * HIDDEN + k0 + c4);
  };
  auto compute = [&](int buf) {
#pragma unroll
    for (int kk = 0; kk < 8; ++kk) {
      const int col = kk * 4 + 2 * half;
      const v2f a = *(const v2f*)&xs[buf][wm * 16 + l16][col];
      const v2f b0 = *(const v2f*)&bg[buf][wn * 16 + l16][col];
      const v2f b1 = *(const v2f*)&bu[buf][wn * 16 + l16][col];
      cg = wmma4(a, b0, cg);
      cu = wmma4(a, b1, cu);
    }
  };

  constexpr int NT = HIDDEN / 32;
  issue(0, 0);
  issue(1, 32);
  int buf = 0;
  for (int it = 0; it < NT - 2; ++it) {
    const int nb = (buf + 2 >= 3) ? buf - 1 : buf + 2;
    issue(nb, (it + 2) * 32);
    __builtin_prefetch(wg + (size_t)(nBase + r0) * HIDDEN + (it + 3) * 32 + c4, 0, 1);
    __builtin_prefetch(wu + (size_t)(nBase + r0) * HIDDEN + (it + 3) * 32 + c4, 0, 1);
    wait_async_le8();
    __syncthreads();
    compute(buf);
    buf = (buf + 1 >= 3) ? 0 : buf + 1;
  }
  wait_async_le4();
  __syncthreads();
  compute(buf);
  buf = (buf + 1 >= 3) ? 0 : buf + 1;
  wait_async_0();
  __syncthreads();
  compute(buf);

#pragma unroll
  for (int r = 0; r < 8; ++r) {
    const int m = mBase + wm * 16 + r + 8 * half;
    const int n = nBase + wn * 16 + l16;
    act[(size_t)m * SH_INTER + n] = silu(cg[r]) * cu[r];
  }
}

// ---------------------------------------------------------------------------
// Shared expert, stage 2: out = act @ wd^T   (overwrites d_out)
// ---------------------------------------------------------------------------
__global__ __launch_bounds__(256) void shared_mlp2_kernel(
    const float* __restrict__ act, const float* __restrict__ wd,
    float* __restrict__ out) {
  __shared__ float xs[3][64][36];
  __shared__ float bs[3][32][36];
  const int tid = threadIdx.x;
  const int lane = tid & 31, wv = tid >> 5;
  const int wm = wv & 3, wn = wv >> 2;
  const int half = lane >> 4, l16 = lane & 15;
  const int mBase = blockIdx.y * 64;
  const int nBase = blockIdx.x * 32;
  const int r0 = tid >> 3;
  const int c4 = (tid & 7) * 4;

  v8f c = {};

  auto issue = [&](int buf, int k0) {
    async_copy16(lds_off(&xs[buf][r0][c4]),
                 act + (size_t)(mBase + r0) * SH_INTER + k0 + c4);
    async_copy16(lds_off(&xs[buf][r0 + 32][c4]),
                 act + (size_t)(mBase + r0 + 32) * SH_INTER + k0 + c4);
    async_copy16(lds_off(&bs[buf][r0][c4]),
                 wd + (size_t)(nBase + r0) * SH_INTER + k0 + c4);
  };
  auto compute = [&](int buf) {
#pragma unroll
    for (int kk = 0; kk < 8; ++kk) {
      const int col = kk * 4 + 2 * half;
      const v2f a = *(const v2f*)&xs[buf][wm * 16 + l16][col];
      const v2f b = *(const v2f*)&bs[buf][wn * 16 + l16][col];
      c = wmma4(a, b, c);
    }
  };

  constexpr int NT = SH_INTER / 32;
  issue(0, 0);
  issue(1, 32);
  int buf = 0;
  for (int it = 0; it < NT - 2; ++it) {
    const int nb = (buf + 2 >= 3) ? buf - 1 : buf + 2;
    issue(nb, (it + 2) * 32);
    __builtin_prefetch(wd + (size_t)(nBase + r0) * SH_INTER + (it + 3) * 32 + c4, 0, 1);
    wait_async_le6();
    __syncthreads();
    compute(buf);
    buf = (buf + 1 >= 3) ? 0 : buf + 1;
  }
  wait_async_le3();
  __syncthreads();
  compute(buf);
  buf = (buf + 1 >= 3) ? 0 : buf + 1;
  wait_async_0();
  __syncthreads();
  compute(buf);

#pragma unroll
  for (int r = 0; r < 8; ++r) {
    const int m = mBase + wm * 16 + r + 8 * half;
    const int n = nBase + wn * 16 + l16;
    out[(size_t)m * HIDDEN + n] = c[r];
  }
}

// ---------------------------------------------------------------------------
// Routed experts, stage 1: act = silu(x_g @ w1e^T) * (x_g @ w3e^T) * route_w
// Tokens gathered per expert; compact act storage: row = offs[e] + slot.
// grid = (INTER/32, TOKENS/64, NEXP); early exit on empty slot tiles.
// ---------------------------------------------------------------------------
__global__ __launch_bounds__(256) void routed_mlp1_kernel(
    const float* __restrict__ x, const float* __restrict__ w1,
    const float* __restrict__ w3, const int* __restrict__ list,
    const float* __restrict__ wgt, const int* __restrict__ counts,
    const int* __restrict__ offs, float* __restrict__ act) {
  const int e = blockIdx.z;
  const int cnt = counts[e];
  const int sBase = blockIdx.y * 64;
  if (sBase >= cnt) return;
  const int iBase = blockIdx.x * 32;

  __shared__ int toks[64];
  __shared__ float xs[3][64][36];
  __shared__ float b1[3][32][36];
  __shared__ float b3[3][32][36];

  const int tid = threadIdx.x;
  if (tid < 64) {
    int s = sBase + tid;
    if (s > cnt - 1) s = cnt - 1;
    toks[tid] = list[e * TOKENS + s];
  }
  __syncthreads();

  const int lane = tid & 31, wv = tid >> 5;
  const int wm = wv & 3, wn = wv >> 2;
  const int half = lane >> 4, l16 = lane & 15;
  const int r0 = tid >> 3;
  const int c4 = (tid & 7) * 4;

  const float* w1e = w1 + (size_t)e * INTER * HIDDEN;
  const float* w3e = w3 + (size_t)e * INTER * HIDDEN;

  v8f cg = {};
  v8f cu = {};

  auto issue = [&](int buf, int k0) {
    async_copy16(lds_off(&xs[buf][r0][c4]),
                 x + (size_t)toks[r0] * HIDDEN + k0 + c4);
    async_copy16(lds_off(&xs[buf][r0 + 32][c4]),
                 x + (size_t)toks[r0 + 32] * HIDDEN + k0 + c4);
    async_copy16(lds_off(&b1[buf][r0][c4]),
                 w1e + (size_t)(iBase + r0) * HIDDEN + k0 + c4);
    async_copy16(lds_off(&b3[buf][r0][c4]),
                 w3e + (size_t)(iBase + r0) * HIDDEN + k0 + c4);
  };
  auto compute = [&](int buf) {
#pragma unroll
    for (int kk = 0; kk < 8; ++kk) {
      const int col = kk * 4 + 2 * half;
      const v2f a = *(const v2f*)&xs[buf][wm * 16 + l16][col];
      const v2f bb1 = *(const v2f*)&b1[buf][wn * 16 + l16][col];
      const v2f bb3 = *(const v2f*)&b3[buf][wn * 16 + l16][col];
      cg = wmma4(a, bb1, cg);
      cu = wmma4(a, bb3, cu);
    }
  };

  constexpr int NT = HIDDEN / 32;
  issue(0, 0);
  issue(1, 32);
  int buf = 0;
  for (int it = 0; it < NT - 2; ++it) {
    const int nb = (buf + 2 >= 3) ? buf - 1 : buf + 2;
    issue(nb, (it + 2) * 32);
    __builtin_prefetch(w1e + (size_t)(iBase + r0) * HIDDEN + (it + 3) * 32 + c4, 0, 1);
    __builtin_prefetch(w3e + (size_t)(iBase + r0) * HIDDEN + (it + 3) * 32 + c4, 0, 1);
    wait_async_le8();
    __syncthreads();
    compute(buf);
    buf = (buf + 1 >= 3) ? 0 : buf + 1;
  }
  wait_async_le4();
  __syncthreads();
  compute(buf);
  buf = (buf + 1 >= 3) ? 0 : buf + 1;
  wait_async_0();
  __syncthreads();
  compute(buf);

  const int off = offs[e];
#pragma unroll
  for (int r = 0; r < 8; ++r) {
    const int slot = sBase + wm * 16 + r + 8 * half;
    if (slot < cnt) {
      const float rw = wgt[e * TOKENS + slot];
      const int n = iBase + wn * 16 + l16;
      act[(size_t)(off + slot) * INTER + n] = silu(cg[r]) * cu[r] * rw;
    }
  }
}

// ---------------------------------------------------------------------------
// Routed experts, stage 2: out[t] += act_pair @ w2e^T   (atomic accumulate)
// grid = (HIDDEN/32, TOKENS/64, NEXP)
// ---------------------------------------------------------------------------
__global__ __launch_bounds__(256) void routed_mlp2_kernel(
    const float* __restrict__ act, const float* __restrict__ w2,
    const int* __restrict__ list, const int* __restrict__ counts,
    const int* __restrict__ offs, float* __restrict__ out) {
  const int e = blockIdx.z;
  const int cnt = counts[e];
  const int sBase = blockIdx.y * 64;
  if (sBase >= cnt) return;
  const int hBase = blockIdx.x * 32;
  const int off = offs[e];

  __shared__ float xs[3][64][36];
  __shared__ float bs[3][32][36];

  const int tid = threadIdx.x;
  const int lane = tid & 31, wv = tid >> 5;
  const int wm = wv & 3, wn = wv >> 2;
  const int half = lane >> 4, l16 = lane & 15;
  const int r0 = tid >> 3;
  const int c4 = (tid & 7) * 4;

  const float* w2e = w2 + (size_t)e * HIDDEN * INTER;

  int sA = sBase + r0;
  if (sA > cnt - 1) sA = cnt - 1;
  int sB = sBase + r0 + 32;
  if (sB > cnt - 1) sB = cnt - 1;
  const size_t rowA = (size_t)(off + sA) * INTER;
  const size_t rowB = (size_t)(off + sB) * INTER;

  v8f c = {};

  auto issue = [&](int buf, int k0) {
    async_copy16(lds_off(&xs[buf][r0][c4]), act + rowA + k0 + c4);
    async_copy16(lds_off(&xs[buf][r0 + 32][c4]), act + rowB + k0 + c4);
    async_copy16(lds_off(&bs[buf][r0][c4]),
                 w2e + (size_t)(hBase + r0) * INTER + k0 + c4);
  };
  auto compute = [&](int buf) {
#pragma unroll
    for (int kk = 0; kk < 8; ++kk) {
      const int col = kk * 4 + 2 * half;
      const v2f a = *(const v2f*)&xs[buf][wm * 16 + l16][col];
      const v2f b = *(const v2f*)&bs[buf][wn * 16 + l16][col];
      c = wmma4(a, b, c);
    }
  };

  constexpr int NT = INTER / 32;
  issue(0, 0);
  issue(1, 32);
  int buf = 0;
  for (int it = 0; it < NT - 2; ++it) {
    const int nb = (buf + 2 >= 3) ? buf - 1 : buf + 2;
    issue(nb, (it + 2) * 32);
    __builtin_prefetch(w2e + (size_t)(hBase + r0) * INTER + (it + 3) * 32 + c4, 0, 1);
    wait_async_le6();
    __syncthreads();
    compute(buf);
    buf = (buf + 1 >= 3) ? 0 : buf + 1;
  }
  wait_async_le3();
  __syncthreads();
  compute(buf);
  buf = (buf + 1 >= 3) ? 0 : buf + 1;
  wait_async_0();
  __syncthreads();
  compute(buf);

#pragma unroll
  for (int r = 0; r < 8; ++r) {
    const int slot = sBase + wm * 16 + r + 8 * half;
    if (slot < cnt) {
      const int t = list[e * TOKENS + slot];
      const int n = hBase + wn * 16 + l16;
      atomicAdd(&out[(size_t)t * HIDDEN + n], c[r]);
    }
  }
}

// ---------------------------------------------------------------------------
// Launch
// ---------------------------------------------------------------------------
extern "C" void kernel_launch(void* const* d_in, const int* in_sizes, int n_in,
                              void* d_out, int out_size, void* d_ws,
                              size_t ws_size, hipStream_t stream) {
  const float* x    = (const float*)d_in[0];
  const float* gw   = (const float*)d_in[1];
  const float* gb   = (const float*)d_in[2];
  const float* shwg = (const float*)d_in[3];
  const float* shwu = (const float*)d_in[4];
  const float* shwd = (const float*)d_in[5];
  const float* ew1  = (const float*)d_in[6];
  const float* ew3  = (const float*)d_in[7];
  const float* ew2  = (const float*)d_in[8];
  float* out = (float*)d_out;

  char* wsB = (char*)d_ws;
  float* route  = (float*)(wsB);               // 512*32 f32   = 64 KB
  int*   list   = (int*)(wsB + 0x10000);       // 32*512 i32   = 64 KB
  float* wgt    = (float*)(wsB + 0x20000);     // 32*512 f32   = 64 KB
  int*   counts = (int*)(wsB + 0x30000);       // 32 i32
  int*   offs   = (int*)(wsB + 0x30100);       // 32 i32
  float* act_sh = (float*)(wsB + 0x40000);     // 512*2048 f32 = 4 MB
  float* act_rt = (float*)(wsB + 0x440000);    // 4096*1024 f32 = 16 MB

  gate_kernel<<<TOKENS, 256, 0, stream>>>(x, gw, gb, route);
  compact_kernel<<<1, 64, 0, stream>>>(route, list, wgt, counts, offs);
  shared_mlp1_kernel<<<dim3(SH_INTER / 32, TOKENS / 64), 256, 0, stream>>>(
      x, shwg, shwu, act_sh);
  routed_mlp1_kernel<<<dim3(INTER / 32, TOKENS / 64, NEXP), 256, 0, stream>>>(
      x, ew1, ew3, list, wgt, counts, offs, act_rt);
  shared_mlp2_kernel<<<dim3(HIDDEN / 32, TOKENS / 64), 256, 0, stream>>>(
      act_sh, shwd, out);
  routed_mlp2_kernel<<<dim3(HIDDEN / 32, TOKENS / 64, NEXP), 256, 0, stream>>>(
      act_rt, ew2, list, counts, offs, out);
}